// GaussianRasterizer2_43611097924157
// MI455X (gfx1250) — compile-verified
//
#include <hip/hip_runtime.h>
#include <hip/hip_bf16.h>
#include <stdint.h>

// ---------------------------------------------------------------------------
// Gaussian point rasterizer for MI455X (gfx1250).
//   Pass 1: init per-pixel u64 keys to ~0 (16B stores).
//   Pass 2: project N points, atomicMin u64 key = (depthBits<<32)|idx  (L2-resident).
//   Pass 3: per-pixel decode winner, write idx/color/depth; feature rows gathered
//           via TDM gather-mode descriptors (8 x 32-bit row indices per descriptor,
//           32 floats/row) into LDS, then stored fully coalesced.
// CDNA5-specific paths used: tensor_load_to_lds (TDM gather), s_wait_tensorcnt,
// ds_load_b128 readback of TDM-staged data.
// WMMA is structurally inapplicable: there is no matmul in this workload.
// ---------------------------------------------------------------------------

#define HH    1080
#define WW    1920
#define HWPIX (HH * WW)          // 2,073,600 (divisible by 512)
#define NFEAT 32
#define KSH   16
#define SH_C0 0.28209479177387814f
#define ZNEAR 0.2f

typedef unsigned int u32x4 __attribute__((ext_vector_type(4)));
typedef int          i32x8 __attribute__((ext_vector_type(8)));
typedef int          i32x4 __attribute__((ext_vector_type(4)));
typedef float        f32x4 __attribute__((ext_vector_type(4)));

#define LDSPTR(T) __attribute__((address_space(3))) T

#if __has_builtin(__builtin_amdgcn_tensor_load_to_lds) && __has_builtin(__builtin_amdgcn_s_wait_tensorcnt)
#define USE_TDM 1
#endif

// Generic pointer -> LDS byte offset (addrspace(3) pointers are 32-bit offsets).
__device__ __forceinline__ unsigned lds_byte_offset(const void* p) {
  return (unsigned)(unsigned long long)(LDSPTR(const char)*)p;
}

// -------------------------------- Pass 1 -----------------------------------
// Two u64 keys per lane via one b128 store.
__global__ void k_init_keys(uint4* __restrict__ keys4) {
  int i = blockIdx.x * blockDim.x + threadIdx.x;
  if (i < HWPIX / 2) {
    keys4[i] = make_uint4(0xFFFFFFFFu, 0xFFFFFFFFu, 0xFFFFFFFFu, 0xFFFFFFFFu);
  }
}

// -------------------------------- Pass 2 -----------------------------------
__global__ void k_scatter(const float* __restrict__ means3D,
                          const float* __restrict__ viewm,
                          const float* __restrict__ projm,
                          unsigned long long* __restrict__ keys, int n) {
  int i = blockIdx.x * blockDim.x + threadIdx.x;
  if (i >= n) return;
  float x = means3D[3 * i + 0];
  float y = means3D[3 * i + 1];
  float z = means3D[3 * i + 2];
  // ph = [x y z 1] @ projm (projm is row-major as passed)
  float p0 = fmaf(x, projm[0], fmaf(y, projm[4], fmaf(z, projm[8],  projm[12])));
  float p1 = fmaf(x, projm[1], fmaf(y, projm[5], fmaf(z, projm[9],  projm[13])));
  float p3 = fmaf(x, projm[3], fmaf(y, projm[7], fmaf(z, projm[11], projm[15])));
  // depth = ([x y z 1] @ viewm)[2]
  float depth = fmaf(x, viewm[2], fmaf(y, viewm[6], fmaf(z, viewm[10], viewm[14])));
  float invw = 1.0f / (p3 + 1e-7f);
  int px = __float2int_rn(fmaf(p0 * invw + 1.0f, (float)WW * 0.5f, -0.5f));
  int py = __float2int_rn(fmaf(p1 * invw + 1.0f, (float)HH * 0.5f, -0.5f));
  if (depth > ZNEAR && px >= 0 && px < WW && py >= 0 && py < HH) {
    unsigned long long key =
        ((unsigned long long)__float_as_uint(depth) << 32) | (unsigned)i;
    atomicMin(&keys[py * WW + px], key);   // resolves in 192MB L2 (16.6MB buffer)
  }
}

// -------------------------------- Pass 3 -----------------------------------
// Block = 256 threads = 8 waves; each wave owns 32 consecutive pixels and a
// 4KB LDS slab (32 rows x 128B). TDM gathers the 32 feature rows into LDS
// (4 descriptors x 8 rows, descriptor q at slab base + q*4KB), then the wave writes the contiguous 4KB output.
__global__ __launch_bounds__(256) void
k_gather(const unsigned long long* __restrict__ keys,
         const float* __restrict__ shs,
         const float* __restrict__ feat,
         const float* __restrict__ bg,
         float* __restrict__ out, int n) {
  __shared__ float smem[8 * 32 * NFEAT];   // 8 waves * 4KB = 32KB
  const int t    = threadIdx.x;
  const int lane = t & 31;
  const int wave = t >> 5;
  const int pix  = blockIdx.x * 256 + t;

  unsigned long long key = keys[pix];
  unsigned idx = (unsigned)key;
  bool hit = idx < (unsigned)n;            // unwritten pixels decode to 0xFFFFFFFF
  int idxc = hit ? (int)idx : 0;

  // idxmap (int32 bits) and depthmap
  ((int*)out)[pix] = hit ? (int)idx : -1;
  out[(size_t)HWPIX * 4 + pix] = hit ? __uint_as_float((unsigned)(key >> 32)) : 0.0f;

  // colmap: DC SH term
  float r, g, b;
  if (hit) {
    const float* sp = shs + (size_t)idxc * (KSH * 3);
    r = fmaxf(fmaf(SH_C0, sp[0], 0.5f), 0.0f);
    g = fmaxf(fmaf(SH_C0, sp[1], 0.5f), 0.0f);
    b = fmaxf(fmaf(SH_C0, sp[2], 0.5f), 0.0f);
  } else {
    r = bg[0]; g = bg[1]; b = bg[2];
  }
  out[(size_t)HWPIX + 3 * (size_t)pix + 0] = r;
  out[(size_t)HWPIX + 3 * (size_t)pix + 1] = g;
  out[(size_t)HWPIX + 3 * (size_t)pix + 2] = b;

  unsigned hmask = (unsigned)__ballot(hit);
  float* myLds = &smem[wave * (32 * NFEAT)];

#ifdef USE_TDM
  {
    const unsigned ldsbase = lds_byte_offset(myLds);
    const unsigned long long ga = (unsigned long long)(uintptr_t)feat;
    const i32x8 zpad = {0, 0, 0, 0, 0, 0, 0, 0};
#pragma unroll
    for (int q = 0; q < 4; ++q) {
      u32x4 g0; i32x8 g1; i32x4 g2, g3;
      // Group0: count=1, gather_mode=1, 32-bit indices; lds_addr; global base; type=2
      g0[0] = 0xC0000001u;
      g0[1] = ldsbase + (unsigned)q * 4096u;
      g0[2] = (unsigned)(ga & 0xFFFFFFFFull);
      g0[3] = (unsigned)((ga >> 32) & 0x1FFFFFFull) | (2u << 30);
      // Group1: data_size=4B; tensor_dim0=32; tensor_dim1=n rows; tile_dim0=32;
      //         tile_dim1=8 valid indices; row stride = 32 elements
      g1[0] = (2 << 16);
      g1[1] = (NFEAT << 16);
      g1[2] = ((n & 0xFFFF) << 16);
      g1[3] = ((n >> 16) & 0xFFFF) | (NFEAT << 16);
      g1[4] = 8;
      g1[5] = NFEAT;
      g1[6] = 0;
      g1[7] = 0;
      // Groups 2/3: eight 32-bit row indices (wave-uniform via readlane)
      g2[0] = __builtin_amdgcn_readlane(idxc, 8 * q + 0);
      g2[1] = __builtin_amdgcn_readlane(idxc, 8 * q + 1);
      g2[2] = __builtin_amdgcn_readlane(idxc, 8 * q + 2);
      g2[3] = __builtin_amdgcn_readlane(idxc, 8 * q + 3);
      g3[0] = __builtin_amdgcn_readlane(idxc, 8 * q + 4);
      g3[1] = __builtin_amdgcn_readlane(idxc, 8 * q + 5);
      g3[2] = __builtin_amdgcn_readlane(idxc, 8 * q + 6);
      g3[3] = __builtin_amdgcn_readlane(idxc, 8 * q + 7);
      // 6-arg form (clang-23 / therock headers): groups 0..3, extra v8 (unused,
      // zero-filled like the TDM header does for absent groups), cache policy.
      __builtin_amdgcn_tensor_load_to_lds(g0, g1, g2, g3, zpad, 0);
    }
    __builtin_amdgcn_s_wait_tensorcnt(0);
  }
#else
  {
    // Fallback: each lane stages its own 128B row into LDS with B128 loads.
    const float4* src = (const float4*)feat + (size_t)idxc * 8;
    float4* dst = (float4*)myLds + lane * 8;
#pragma unroll
    for (int j = 0; j < 8; ++j) dst[j] = src[j];
  }
#endif
  __syncthreads();

  // Launder the LDS pointer through an asm operand: the TDM builtin carries no
  // LDS-pointer operand, so without this LLVM sees smem as store-free and folds
  // the readback loads to zero. An asm-defined addrspace(3) pointer is opaque,
  // forcing real ds_load_b128, and keeps the accesses provably LDS.
  LDSPTR(char)* lp = (LDSPTR(char)*)myLds;
  asm volatile("" : "+v"(lp) :: "memory");
  const LDSPTR(const f32x4)* lds4 = (const LDSPTR(const f32x4)*)lp;

  // Coalesced featuremap store: 32 consecutive pixels' rows are one contiguous
  // 4KB block; lane-crossing LDS reads feed 512B-per-instruction global stores.
  const int pix0 = blockIdx.x * 256 + wave * 32;
  f32x4* out4 = (f32x4*)(out + (size_t)HWPIX * 5) + (size_t)pix0 * 8;
  const f32x4 z4 = {0.f, 0.f, 0.f, 0.f};
#pragma unroll
  for (int j = 0; j < 8; ++j) {
    int e = j * 32 + lane;                    // element within the wave's 4KB block
    f32x4 v = ((hmask >> (e >> 3)) & 1u) ? lds4[e] : z4;
    out4[e] = v;
  }
}

// ------------------------------ launcher -----------------------------------
extern "C" void kernel_launch(void* const* d_in, const int* in_sizes, int n_in,
                              void* d_out, int out_size, void* d_ws, size_t ws_size,
                              hipStream_t stream) {
  const float* means3D = (const float*)d_in[0];
  const float* featv   = (const float*)d_in[3];
  const float* shs     = (const float*)d_in[4];
  const float* viewm   = (const float*)d_in[7];
  const float* projm   = (const float*)d_in[8];
  const float* bg      = (const float*)d_in[9];
  const int n = in_sizes[0] / 3;

  unsigned long long* keys = (unsigned long long*)d_ws;  // needs HWPIX*8 = 16.6MB
  float* out = (float*)d_out;

  k_init_keys<<<(HWPIX / 2 + 255) / 256, 256, 0, stream>>>((uint4*)d_ws);
  k_scatter<<<(n + 255) / 256, 256, 0, stream>>>(means3D, viewm, projm, keys, n);
  k_gather<<<HWPIX / 256, 256, 0, stream>>>(keys, shs, featv, bg, out, n);
}